// GAT_53472342835253
// MI455X (gfx1250) — compile-verified
//
#include <hip/hip_runtime.h>
#include <hip/hip_bf16.h>

#define NODES_DIV16_OK 1
#define DIMF 128           // feature dim (both IN and D are 128)
#define SLOPE 0.2f

typedef __attribute__((ext_vector_type(2))) float v2f;
typedef __attribute__((ext_vector_type(8))) float v8f;

// ---------------------------------------------------------------------------
// Init: m = -inf, s = 0, agg = 0
// ---------------------------------------------------------------------------
__global__ void gat_init_kernel(float* __restrict__ m, float* __restrict__ s,
                                float* __restrict__ agg, int n_nodes, long long agg_n) {
    long long i = (long long)blockIdx.x * blockDim.x + threadIdx.x;
    if (i < n_nodes) {
        m[i] = __int_as_float(0xFF800000);   // -inf
        s[i] = 0.0f;
    }
    if (i < agg_n) agg[i] = 0.0f;
}

// ---------------------------------------------------------------------------
// GEMM + bias: Y[N x 128] = X[N x 128] @ W[128 x 128] + b
// Block = 256 threads = 8 waves. blockIdx.x selects a 16-row stripe of X,
// staged once into LDS; wave w computes the 16x16 output tile at cols w*16.
// K-loop: 32 x v_wmma_f32_16x16x4_f32.
// Fragment layouts per CDNA5 ISA 7.12.2:
//   A 16x4 f32 : lanes 0-15 -> M=lane, v0=K0,v1=K1 ; lanes 16-31 -> v0=K2,v1=K3
//   B 4x16 f32 : v0: lanes0-15 = row K0 (N=0..15), lanes16-31 = row K2; v1: K1/K3
//   C/D 16x16  : vr: lanes0-15 -> M=r, lanes16-31 -> M=r+8 ; N = lane&15
// ---------------------------------------------------------------------------
__global__ void __launch_bounds__(256)
gat_gemm_bias_kernel(const float* __restrict__ X, const float* __restrict__ W,
                     const float* __restrict__ b, float* __restrict__ Y) {
    __shared__ float As[16 * DIMF];                 // 8 KB A stripe
    const int tid = threadIdx.x;

    // cooperative load: 16*128 floats = 512 float4; 256 threads x 2
    const float4* Xg = (const float4*)(X + (size_t)blockIdx.x * 16 * DIMF);
    float4* As4 = (float4*)As;
    As4[tid]       = Xg[tid];
    As4[tid + 256] = Xg[tid + 256];
    __syncthreads();

    const int wv   = tid >> 5;
    const int lane = tid & 31;
    const int half = lane >> 4;      // 0 or 1
    const int lr   = lane & 15;
    const int colbase = wv * 16;

    v8f c = {};
    #pragma unroll
    for (int k0 = 0; k0 < DIMF; k0 += 4) {
        const int ka = k0 + 2 * half;
        v2f a, bm;
        a.x  = As[lr * DIMF + ka];
        a.y  = As[lr * DIMF + ka + 1];
        bm.x = W[(size_t)ka * DIMF + colbase + lr];
        bm.y = W[(size_t)(ka + 1) * DIMF + colbase + lr];
        c = __builtin_amdgcn_wmma_f32_16x16x4_f32(false, a, false, bm,
                                                  (short)0, c, false, false);
    }

    const float bb = b[colbase + lr];
    const size_t rowbase = (size_t)blockIdx.x * 16;
    #pragma unroll
    for (int r = 0; r < 8; ++r) {
        const int mrow = r + 8 * half;
        Y[(rowbase + mrow) * DIMF + colbase + lr] = c[r] + bb;
    }
}

// ---------------------------------------------------------------------------
// Float atomic max via int atomics (native VMEM atomics, correct for all signs
// given -inf initialization).
// ---------------------------------------------------------------------------
__device__ __forceinline__ void atomicMaxF(float* addr, float v) {
    if (v >= 0.0f) atomicMax((int*)addr, __float_as_int(v));
    else           atomicMin((unsigned int*)addr, __float_as_uint(v));
}

// ---------------------------------------------------------------------------
// Edge pass 1: logits[e] = sum_d lrelu(xl[src][d] + xr[dst][d]) * att[d]
//              m[dst] = max(m[dst], logits[e])
// One wave per edge; lane i handles features 4i..4i+3 as float4 (coalesced
// 512B row reads that stay resident in L2).
// ---------------------------------------------------------------------------
__global__ void __launch_bounds__(256)
gat_edge_logits_kernel(const float* __restrict__ xl, const float* __restrict__ xr,
                       const int* __restrict__ ei, const float* __restrict__ att,
                       float* __restrict__ logits, float* __restrict__ m, int E_) {
    const int wid  = threadIdx.x >> 5;
    const int lane = threadIdx.x & 31;
    const int e = blockIdx.x * 8 + wid;
    if (e >= E_) return;
    const int src = ei[e];
    const int dst = ei[E_ + e];

    const float4 av = ((const float4*)att)[lane];
    const float4 l  = ((const float4*)(xl + (size_t)src * DIMF))[lane];
    const float4 r  = ((const float4*)(xr + (size_t)dst * DIMF))[lane];

    float acc = 0.0f, t;
    t = l.x + r.x; acc += (t > 0.0f ? t : SLOPE * t) * av.x;
    t = l.y + r.y; acc += (t > 0.0f ? t : SLOPE * t) * av.y;
    t = l.z + r.z; acc += (t > 0.0f ? t : SLOPE * t) * av.z;
    t = l.w + r.w; acc += (t > 0.0f ? t : SLOPE * t) * av.w;

    #pragma unroll
    for (int o = 16; o > 0; o >>= 1) acc += __shfl_xor(acc, o, 32);

    if (lane == 0) {
        logits[e] = acc;
        atomicMaxF(&m[dst], acc);
    }
}

// ---------------------------------------------------------------------------
// Edge pass 2: a = exp(logit - m[dst]); s[dst] += a   (a overwrites logits)
// ---------------------------------------------------------------------------
__global__ void gat_edge_exp_kernel(float* __restrict__ logits, const float* __restrict__ m,
                                    const int* __restrict__ ei, float* __restrict__ s, int E_) {
    const int e = blockIdx.x * blockDim.x + threadIdx.x;
    if (e >= E_) return;
    const int dst = ei[E_ + e];
    const float a = __expf(logits[e] - m[dst]);
    logits[e] = a;
    unsafeAtomicAdd(&s[dst], a);
}

// ---------------------------------------------------------------------------
// Edge pass 3: agg[dst] += (a[e]/max(s[dst],1e-16)) * xl[src]
// One wave per edge; 4 native f32 atomic adds per lane (global_atomic_add_f32).
// ---------------------------------------------------------------------------
__global__ void __launch_bounds__(256)
gat_edge_aggregate_kernel(const float* __restrict__ xl, const float* __restrict__ a,
                          const float* __restrict__ s, const int* __restrict__ ei,
                          float* __restrict__ agg, int E_) {
    const int wid  = threadIdx.x >> 5;
    const int lane = threadIdx.x & 31;
    const int e = blockIdx.x * 8 + wid;
    if (e >= E_) return;
    const int src = ei[e];
    const int dst = ei[E_ + e];

    const float coef = a[e] / fmaxf(s[dst], 1e-16f);
    const float4 v = ((const float4*)(xl + (size_t)src * DIMF))[lane];
    float* base = agg + (size_t)dst * DIMF + lane * 4;
    unsafeAtomicAdd(base + 0, coef * v.x);
    unsafeAtomicAdd(base + 1, coef * v.y);
    unsafeAtomicAdd(base + 2, coef * v.z);
    unsafeAtomicAdd(base + 3, coef * v.w);
}

// ---------------------------------------------------------------------------
// out = relu(agg + bias)
// ---------------------------------------------------------------------------
__global__ void gat_bias_relu_kernel(const float* __restrict__ agg,
                                     const float* __restrict__ bias,
                                     float* __restrict__ out, long long n) {
    long long i = (long long)blockIdx.x * blockDim.x + threadIdx.x;
    if (i >= n) return;
    out[i] = fmaxf(agg[i] + bias[i & (DIMF - 1)], 0.0f);
}

// ---------------------------------------------------------------------------
extern "C" void kernel_launch(void* const* d_in, const int* in_sizes, int n_in,
                              void* d_out, int out_size, void* d_ws, size_t ws_size,
                              hipStream_t stream) {
    const float* x0    = (const float*)d_in[0];
    const int*   ei    = (const int*)d_in[1];
    const float* Wl1   = (const float*)d_in[2];
    const float* bl1   = (const float*)d_in[3];
    const float* Wr1   = (const float*)d_in[4];
    const float* br1   = (const float*)d_in[5];
    const float* att1  = (const float*)d_in[6];
    const float* bias1 = (const float*)d_in[7];
    const float* Wl2   = (const float*)d_in[8];
    const float* bl2   = (const float*)d_in[9];
    const float* Wr2   = (const float*)d_in[10];
    const float* br2   = (const float*)d_in[11];
    const float* att2  = (const float*)d_in[12];
    const float* bias2 = (const float*)d_in[13];

    const int Nn = in_sizes[0] / DIMF;   // 50000
    const int Ee = in_sizes[1] / 2;      // 800000
    const long long aggN = (long long)Nn * DIMF;

    char* ws = (char*)d_ws;
    const size_t NB = (size_t)Nn * DIMF * sizeof(float);           // 25.6 MB
    const size_t EB = (((size_t)Ee * sizeof(float)) + 255) & ~(size_t)255;
    const size_t MB = (((size_t)Nn * sizeof(float)) + 255) & ~(size_t)255;
    float* xl     = (float*)(ws);
    float* xr     = (float*)(ws + NB);
    float* agg    = (float*)(ws + 2 * NB);
    float* x1     = (float*)(ws + 3 * NB);
    float* logits = (float*)(ws + 4 * NB);
    float* mbuf   = (float*)(ws + 4 * NB + EB);
    float* sbuf   = (float*)(ws + 4 * NB + EB + MB);

    const int gemm_blocks = Nn / 16;                 // 3125
    const int edge_wblocks = (Ee + 7) / 8;           // wave-per-edge
    const int edge_tblocks = (Ee + 255) / 256;
    const int init_blocks = (int)((aggN + 255) / 256);
    const int elem_blocks = init_blocks;

    // ---------------- layer 1 ----------------
    gat_init_kernel<<<init_blocks, 256, 0, stream>>>(mbuf, sbuf, agg, Nn, aggN);
    gat_gemm_bias_kernel<<<gemm_blocks, 256, 0, stream>>>(x0, Wl1, bl1, xl);
    gat_gemm_bias_kernel<<<gemm_blocks, 256, 0, stream>>>(x0, Wr1, br1, xr);
    gat_edge_logits_kernel<<<edge_wblocks, 256, 0, stream>>>(xl, xr, ei, att1, logits, mbuf, Ee);
    gat_edge_exp_kernel<<<edge_tblocks, 256, 0, stream>>>(logits, mbuf, ei, sbuf, Ee);
    gat_edge_aggregate_kernel<<<edge_wblocks, 256, 0, stream>>>(xl, logits, sbuf, ei, agg, Ee);
    gat_bias_relu_kernel<<<elem_blocks, 256, 0, stream>>>(agg, bias1, x1, aggN);

    // ---------------- layer 2 ----------------
    gat_init_kernel<<<init_blocks, 256, 0, stream>>>(mbuf, sbuf, agg, Nn, aggN);
    gat_gemm_bias_kernel<<<gemm_blocks, 256, 0, stream>>>(x1, Wl2, bl2, xl);
    gat_gemm_bias_kernel<<<gemm_blocks, 256, 0, stream>>>(x1, Wr2, br2, xr);
    gat_edge_logits_kernel<<<edge_wblocks, 256, 0, stream>>>(xl, xr, ei, att2, logits, mbuf, Ee);
    gat_edge_exp_kernel<<<edge_tblocks, 256, 0, stream>>>(logits, mbuf, ei, sbuf, Ee);
    gat_edge_aggregate_kernel<<<edge_wblocks, 256, 0, stream>>>(xl, logits, sbuf, ei, agg, Ee);
    gat_bias_relu_kernel<<<elem_blocks, 256, 0, stream>>>(agg, bias2, (float*)d_out, aggN);
}